// GCN_53163105189936
// MI455X (gfx1250) — compile-verified
//
#include <hip/hip_runtime.h>
#include <hip/hip_bf16.h>

typedef float v2f __attribute__((ext_vector_type(2)));
typedef float v8f __attribute__((ext_vector_type(8)));

static __device__ __forceinline__ void atomic_add_f32(float* p, float v) {
    // lowers to global_atomic_add_f32 (no CAS loop) on gfx1250
    unsafeAtomicAdd(p, v);
}

// -------- fill buffer with broadcast bias row (float4 granularity) --------
// p[i] = b4[i & dmask4]  where dmask4 = D/4 - 1 (D power of two, rows aligned)
__global__ void bias_fill_kernel(float4* __restrict__ p, const float4* __restrict__ b4,
                                 int dmask4, long long n4) {
    long long i = (long long)blockIdx.x * blockDim.x + threadIdx.x;
    long long stride = (long long)gridDim.x * blockDim.x;
    for (; i < n4; i += stride) p[i] = b4[(int)(i & dmask4)];
}

// ---------------- zero fill (float4 granularity) ----------------
__global__ void zero_kernel(float4* __restrict__ p, long long n4) {
    long long i = (long long)blockIdx.x * blockDim.x + threadIdx.x;
    long long stride = (long long)gridDim.x * blockDim.x;
    float4 z; z.x = 0.f; z.y = 0.f; z.z = 0.f; z.w = 0.f;
    for (; i < n4; i += stride) p[i] = z;
}

// ---------------- degree: deg[col[e]] += w[e] ----------------
__global__ void deg_kernel(const int* __restrict__ col, const float* __restrict__ w,
                           float* __restrict__ deg, int E) {
    int i = blockIdx.x * blockDim.x + threadIdx.x;
    if (i < E) atomic_add_f32(&deg[col[i]], w[i]);
}

// ---------------- dis = deg>0 ? rsqrt(deg) : 0  (in place) ----------------
__global__ void dis_kernel(float* __restrict__ deg, int N) {
    int i = blockIdx.x * blockDim.x + threadIdx.x;
    if (i < N) {
        float d = deg[i];
        deg[i] = (d > 0.f) ? rsqrtf(d) : 0.f;
    }
}

// ---------------- norm[e] = dis[row]*w*dis[col] ----------------
__global__ void norm_kernel(const int* __restrict__ row, const int* __restrict__ col,
                            const float* __restrict__ w, const float* __restrict__ dis,
                            float* __restrict__ nrm, int E) {
    int i = blockIdx.x * blockDim.x + threadIdx.x;
    if (i < E) nrm[i] = dis[row[i]] * w[i] * dis[col[i]];
}

// ---------------- fp32 GEMM via V_WMMA_F32_16X16X4_F32 ----------------
// C[N x M] = op(A)[N x K] @ B[K x M], row-major, N%16==0, M%16==0, K%4==0.
// RELU_A applies max(a,0) to A elements on load (fuses previous layer's ReLU).
// One wave computes one 16x16 tile of C. blockDim.x = 128 (4 waves).
template <bool RELU_A>
__global__ void gemm_wmma_f32(const float* __restrict__ A, const float* __restrict__ B,
                              float* __restrict__ C, int N, int K, int M) {
    const int wavesPerBlock = blockDim.x >> 5;
    const int wave = blockIdx.x * wavesPerBlock + (threadIdx.x >> 5);
    const int lane = threadIdx.x & 31;
    const int tilesPerRow = M >> 4;
    const int totalTiles = (N >> 4) * tilesPerRow;
    if (wave >= totalTiles) return;           // wave-uniform: EXEC stays all-1s
    const int tm = wave / tilesPerRow;
    const int tn = wave % tilesPerRow;
    const int hi = lane >> 4;                 // 0: lanes 0-15, 1: lanes 16-31
    const int l  = lane & 15;

    // A lane base: row (tm*16+l), K offset hi*2 (K pairs contiguous -> float2 loads)
    const float* Arow = A + (size_t)(tm * 16 + l) * K + hi * 2;
    // B lane base: col (tn*16+l), K offset hi*2
    const float* Bcol = B + (size_t)(hi * 2) * M + tn * 16 + l;

    v8f acc = {};
    for (int k0 = 0; k0 < K; k0 += 4) {
        v2f a = *(const v2f*)(Arow + k0);     // K = k0+hi*2, k0+1+hi*2
        if (RELU_A) {
            a.x = fmaxf(a.x, 0.f);
            a.y = fmaxf(a.y, 0.f);
        }
        v2f b;
        b.x = Bcol[(size_t)k0 * M];           // K = k0+hi*2
        b.y = Bcol[(size_t)(k0 + 1) * M];     // K = k0+1+hi*2
        acc = __builtin_amdgcn_wmma_f32_16x16x4_f32(
            /*neg_a=*/false, a, /*neg_b=*/false, b,
            /*c_mod=*/(short)0, acc, /*reuse_a=*/false, /*reuse_b=*/false);
    }

    // C layout: VGPR r -> rows (tm*16 + r + hi*8), col tn*16 + l
    float* Crow = C + (size_t)(tm * 16 + hi * 8) * M + tn * 16 + l;
#pragma unroll
    for (int r = 0; r < 8; ++r) {
        Crow[(size_t)r * M] = acc[r];
    }
}

// ---------------- edge scatter: out[col] += norm * xw[row] ----------------
// D/4 lanes per edge; each lane handles 4 consecutive features (float4 gather).
// lshift = log2(D/4): 5 for D=128, 4 for D=64. total = E << lshift.
__global__ void scatter_kernel(const float* __restrict__ xw, const float* __restrict__ nrm,
                               const int* __restrict__ row, const int* __restrict__ col,
                               float* __restrict__ out, long long total, int D, int lshift) {
    long long gid = (long long)blockIdx.x * blockDim.x + threadIdx.x;
    if (gid >= total) return;
    int e    = (int)(gid >> lshift);
    int lane = (int)(gid & ((1 << lshift) - 1));
    int f0 = lane * 4;
    int r = row[e];
    int c = col[e];
    float w = nrm[e];
    const float4 v = *(const float4*)(xw + (size_t)r * D + f0);
    float* dst = out + (size_t)c * D + f0;
    atomic_add_f32(dst + 0, v.x * w);
    atomic_add_f32(dst + 1, v.y * w);
    atomic_add_f32(dst + 2, v.z * w);
    atomic_add_f32(dst + 3, v.w * w);
}

static inline int cdiv(long long a, long long b) { return (int)((a + b - 1) / b); }

extern "C" void kernel_launch(void* const* d_in, const int* in_sizes, int n_in,
                              void* d_out, int out_size, void* d_ws, size_t ws_size,
                              hipStream_t stream) {
    const float* x  = (const float*)d_in[0];
    const int*   ei = (const int*)d_in[1];
    const float* ew = (const float*)d_in[2];
    const float* W0 = (const float*)d_in[3];
    const float* b0 = (const float*)d_in[4];
    const float* W1 = (const float*)d_in[5];
    const float* b1 = (const float*)d_in[6];
    const float* W2 = (const float*)d_in[7];
    const float* b2 = (const float*)d_in[8];

    const int DIN = 128, DH = 128, DOUT = 64;
    const int N = in_sizes[0] / DIN;
    const int E = in_sizes[2];
    const int* row = ei;        // edge_index[0]
    const int* col = ei + E;    // edge_index[1]

    // workspace layout (floats)
    const size_t degPad = ((size_t)N + 1023) & ~(size_t)1023;
    const size_t nrmPad = ((size_t)E + 1023) & ~(size_t)1023;
    float* deg  = (float*)d_ws;
    float* nrm  = deg + degPad;
    float* bufA = nrm + nrmPad;                // xw scratch, N x 128
    float* bufB = bufA + (size_t)N * DH;       // aggregation (agg + bias), N x 128
    float* out  = (float*)d_out;               // N x 64

    const int B256 = 256;

    // ---- gcn_norm ----
    zero_kernel<<<cdiv((long long)(degPad / 4), B256), B256, 0, stream>>>(
        (float4*)deg, (long long)(degPad / 4));
    deg_kernel<<<cdiv(E, B256), B256, 0, stream>>>(col, ew, deg, E);
    dis_kernel<<<cdiv(N, B256), B256, 0, stream>>>(deg, N);
    norm_kernel<<<cdiv(E, B256), B256, 0, stream>>>(row, col, ew, deg, nrm, E);

    // ---- layer 0: bufB = agg(x @ W0) + b0  (ReLU deferred to next GEMM) ----
    {
        int tiles = (N / 16) * (DH / 16);
        gemm_wmma_f32<false><<<cdiv(tiles, 4), 128, 0, stream>>>(x, W0, bufA, N, DIN, DH);
        bias_fill_kernel<<<cdiv((long long)N * DH / 4, B256), B256, 0, stream>>>(
            (float4*)bufB, (const float4*)b0, DH / 4 - 1, (long long)N * DH / 4);
        long long total = (long long)E << 5;   // 32 lanes/edge (D=128)
        scatter_kernel<<<cdiv(total, B256), B256, 0, stream>>>(
            bufA, nrm, row, col, bufB, total, DH, 5);
    }

    // ---- layer 1: bufB = agg(relu(bufB) @ W1) + b1 ----
    {
        int tiles = (N / 16) * (DH / 16);
        gemm_wmma_f32<true><<<cdiv(tiles, 4), 128, 0, stream>>>(bufB, W1, bufA, N, DH, DH);
        bias_fill_kernel<<<cdiv((long long)N * DH / 4, B256), B256, 0, stream>>>(
            (float4*)bufB, (const float4*)b1, DH / 4 - 1, (long long)N * DH / 4);
        long long total = (long long)E << 5;
        scatter_kernel<<<cdiv(total, B256), B256, 0, stream>>>(
            bufA, nrm, row, col, bufB, total, DH, 5);
    }

    // ---- layer 2: out = agg(relu(bufB) @ W2) + b2 (no relu) ----
    {
        int tiles = (N / 16) * (DOUT / 16);
        gemm_wmma_f32<true><<<cdiv(tiles, 4), 128, 0, stream>>>(bufB, W2, bufA, N, DH, DOUT);
        bias_fill_kernel<<<cdiv((long long)N * DOUT / 4, B256), B256, 0, stream>>>(
            (float4*)out, (const float4*)b2, DOUT / 4 - 1, (long long)N * DOUT / 4);
        long long total = (long long)E << 4;   // 16 lanes/edge (D=64)
        scatter_kernel<<<cdiv(total, B256), B256, 0, stream>>>(
            bufA, nrm, row, col, out, total, DOUT, 4);
    }
}